// LatticeFilter_68118181314695
// MI455X (gfx1250) — compile-verified
//
#include <hip/hip_runtime.h>
#include <stdint.h>

// Levinson step-up recursion (reflection coeffs -> LPC coeffs) for MI455X.
//
//   B=16, p=128, Tf=8192. out[b,t,j], j=0..128, out[...,0]=1.
//   Per (b,t): a[j] += k_i * a[i+1-j], j=1..i+1, k_i = 0.98*rc[b,i,t].
//
// One thread per (b,t). a[0..128] register-resident: the triangular
// butterfly is emitted as straight-line code via template recursion so
// every index is a compile-time constant (guarantees SROA promotion —
// round 1 showed the pragma-unroll version fell back to scratch).
// Input tile staged with global_load_async_to_lds_b128; output staged to
// LDS in global layout and drained with global_store_async_from_lds_b128.

#define NB 16
#define PP 128
#define TF 8192
#define TW 64          // threads per block (2 waves); TF % TW == 0

// ---- straight-line triangular butterfly via template recursion ---------

template <int I, int J, int END>
struct Inner {
    static __device__ __forceinline__ void run(float (&a)[PP + 1], float k) {
        constexpr int JJ = I + 1 - J;          // pair partner, 1..I
        const float lo = a[J], hi = a[JJ];
        a[J]  = __builtin_fmaf(k, hi, lo);
        a[JJ] = __builtin_fmaf(k, lo, hi);     // J==JJ: same value rewritten
        Inner<I, J + 1, END>::run(a, k);
    }
};
template <int I, int END>
struct Inner<I, END, END> {
    static __device__ __forceinline__ void run(float (&)[PP + 1], float) {}
};

template <int I>
struct Outer {
    static __device__ __forceinline__ void run(float (&a)[PP + 1],
                                               const float* __restrict__ kt,
                                               int tid) {
        const float k = 0.98f * kt[I * TW + tid];   // ds_load, imm offset
        Inner<I, 1, (I + 1) / 2 + 1>::run(a, k);    // pairs (j, i+1-j), j,jj in 1..i
        a[I + 1] = k;                               // 0 + k*a[0], a[0]==1
        Outer<I + 1>::run(a, kt, tid);
    }
};
template <>
struct Outer<PP> {
    static __device__ __forceinline__ void run(float (&)[PP + 1],
                                               const float* __restrict__, int) {}
};

// ------------------------------------------------------------------------

__global__ __launch_bounds__(TW) void lpc_stepup_kernel(
    const float* __restrict__ rc, float* __restrict__ out)
{
    // union buffer: phase 1: k-tile [PP rows][TW cols] = 8192 floats (32 KB)
    //               phase 2: out stage [TW][PP+1]      = 8256 floats (33 KB)
    __shared__ __align__(16) float smem[TW * (PP + 1)];

    const int tid = threadIdx.x;
    const int n0  = blockIdx.x * TW;       // flat (b,t) base for this block
    const int b   = n0 / TF;               // uniform per block
    const int t0  = n0 - b * TF;

    const unsigned lbase = (unsigned)(size_t)&smem[0];   // LDS byte offset

    // ---- async stage rc tile: PP rows x (TW*4 B) = 32 KB ---------------
    // per iteration: 64 lanes x 16 B = 1 KB = 4 rows of 256 B
    {
        const uint64_t gbase =
            (uint64_t)(uintptr_t)(rc + ((size_t)b * PP) * TF + t0);
        const int rsub = tid >> 4;               // row within 4-row chunk
        const int cb   = (tid & 15) << 4;        // byte within row
        #pragma unroll
        for (int it = 0; it < PP / 4; ++it) {
            const int row = it * 4 + rsub;
            unsigned goff = (unsigned)(row * (TF * 4) + cb);
            unsigned loff = lbase + (unsigned)(row * (TW * 4) + cb);
            asm volatile("global_load_async_to_lds_b128 %0, %1, %2"
                         :: "v"(loff), "v"(goff), "s"(gbase) : "memory");
        }
        asm volatile("s_wait_asynccnt 0x0" ::: "memory");
    }
    __syncthreads();   // tile loaded by both waves -> visible to all

    // ---- step-up recursion, a[] fully register-resident ----------------
    float a[PP + 1];
    a[0] = 1.0f;
    Outer<0>::run(a, smem, tid);

    __syncthreads();   // done reading k-tile; reuse smem for output staging

    // ---- stage results in GLOBAL layout order in LDS -------------------
    // dword addr = tid*129 + j  -> bank (tid + j) & 63: conflict-free
    #pragma unroll
    for (int j = 0; j <= PP; ++j)
        smem[tid * (PP + 1) + j] = a[j];
    __syncthreads();

    // ---- async drain LDS -> HBM, 16 B per lane, fully coalesced --------
    {
        const uint64_t obase =
            (uint64_t)(uintptr_t)out + (uint64_t)n0 * ((PP + 1) * 4);
        const int nvec = TW * (PP + 1) / 4;          // 2064 float4 chunks
        for (int it = 0; it < (nvec + TW - 1) / TW; ++it) {
            const int idx = it * TW + tid;
            if (idx < nvec) {
                unsigned goff = (unsigned)(idx * 16);
                unsigned loff = lbase + (unsigned)(idx * 16);
                asm volatile("global_store_async_from_lds_b128 %0, %1, %2"
                             :: "v"(goff), "v"(loff), "s"(obase) : "memory");
            }
        }
        // no explicit wait needed: s_endpgm performs implicit wait-idle
    }
}

extern "C" void kernel_launch(void* const* d_in, const int* in_sizes, int n_in,
                              void* d_out, int out_size, void* d_ws, size_t ws_size,
                              hipStream_t stream) {
    (void)in_sizes; (void)n_in; (void)d_ws; (void)ws_size; (void)out_size;
    // d_in[0] = excitation (dead in the reference), d_in[1] = rc (B,p,Tf) fp32
    const float* rc  = (const float*)d_in[1];
    float*       out = (float*)d_out;
    dim3 grid((NB * TF) / TW);   // 2048 blocks
    dim3 block(TW);              // 64 threads = 2 wave32
    lpc_stepup_kernel<<<grid, block, 0, stream>>>(rc, out);
}